// FieldAwareFM_17970143167337
// MI455X (gfx1250) — compile-verified
//
#include <hip/hip_runtime.h>
#include <hip/hip_bf16.h>

typedef __attribute__((ext_vector_type(2))) float v2f;
typedef __attribute__((ext_vector_type(8))) float v8f;

#define NFIELD 17
#define NNUM   23
#define NNUMP  24   // K padded to a multiple of 4 (column 23 zero-filled in LDS)
#define EDIM   16
#define ROWS_PER_BLOCK 128
#define THREADS 256

// cumsum of VOCAB[:-1]
__device__ __constant__ int OFF[NFIELD] = {
    0, 500, 550, 650, 653, 1000653, 1000663, 1000683, 1000783,
    1000785, 1000787, 1001788, 1001792, 1001796, 1001846, 1003846, 1003856};

__device__ __forceinline__ float red16(float v) {
    // sum across a 16-lane group of a wave32 (xor masks < 16 stay in-group)
    v += __shfl_xor(v, 1, 32);
    v += __shfl_xor(v, 2, 32);
    v += __shfl_xor(v, 4, 32);
    v += __shfl_xor(v, 8, 32);
    return v;
}

__global__ __launch_bounds__(THREADS)
void FieldAwareFM_kernel(const int* __restrict__ x_cat,
                         const float* __restrict__ x_num,
                         const float* __restrict__ table,
                         const float* __restrict__ W_num,
                         const float* __restrict__ bias,
                         float* __restrict__ out,
                         int n_rows)
{
    __shared__ int   s_xcat[ROWS_PER_BLOCK * NFIELD];   // 8704 B
    __shared__ float s_xnum[ROWS_PER_BLOCK * NNUMP];    // 12288 B (K padded)
    __shared__ float s_w[EDIM * NNUMP];                 // 1536 B  (K padded)

    const int tid        = threadIdx.x;
    const int block_row0 = blockIdx.x * ROWS_PER_BLOCK;

    // ---- Stage block inputs into LDS (zero-pad K=23 so WMMA loads are branchless) ----
    for (int t = tid; t < ROWS_PER_BLOCK * NFIELD; t += THREADS)
        s_xcat[t] = x_cat[(size_t)block_row0 * NFIELD + t];
    for (int t = tid; t < ROWS_PER_BLOCK * NNUMP; t += THREADS) {
        const int r = t / NNUMP, c0 = t % NNUMP;
        s_xnum[t] = (c0 < NNUM) ? x_num[(size_t)(block_row0 + r) * NNUM + c0] : 0.0f;
    }
    for (int t = tid; t < EDIM * NNUMP; t += THREADS) {
        const int r = t / NNUMP, c0 = t % NNUMP;
        s_w[t] = (c0 < NNUM) ? W_num[r * NNUM + c0] : 0.0f;
    }
    __syncthreads();

    const int wave  = tid >> 5;      // 0..7, each wave owns 16 rows
    const int lane  = tid & 31;
    const int half  = lane >> 4;     // 0 or 1
    const int l16   = lane & 15;
    const int wrow0 = wave * 16;     // first local row of this wave's tile

    // ---- Phase 1: emb_num(16x16) = x_num(16x23) @ W_num^T(23x16) via WMMA ----
    // V_WMMA_F32_16X16X4_F32, K padded to 24 = 6 chunks of 4; zero pad column
    // makes every operand load unconditional (no EXEC manipulation near WMMA).
    v8f c = {};
#pragma unroll
    for (int kc = 0; kc < 6; ++kc) {
        const int k = kc * 4 + half * 2;  // A: lanes0-15 K=k,k+1; lanes16-31 K=k+2,k+3
        v2f a, b;
        a.x = s_xnum[(wrow0 + l16) * NNUMP + k    ];
        a.y = s_xnum[(wrow0 + l16) * NNUMP + k + 1];
        // B[k][n] = W_num[n][k]; same K striping, N = l16
        b.x = s_w[l16 * NNUMP + k    ];
        b.y = s_w[l16 * NNUMP + k + 1];
        c = __builtin_amdgcn_wmma_f32_16x16x4_f32(
                /*neg_a=*/false, a, /*neg_b=*/false, b,
                /*c_mod=*/(short)0, c, /*reuse_a=*/false, /*reuse_b=*/false);
    }
    // Per-row sum over the 16 embedding dims: reduce C across each 16-lane group.
    // C layout: VGPR j -> M=j (lanes 0-15), M=j+8 (lanes 16-31), N=lane%16.
    float embsum[8];
#pragma unroll
    for (int j = 0; j < 8; ++j)
        embsum[j] = red16(c[j]);   // lanes0-15: rowsum(wrow0+j); lanes16-31: rowsum(wrow0+8+j)

    const float b0 = bias[0];

    // ---- Phase 2: embedding gathers + FM reduction ----
    // Iteration j: lanes0-15 handle local row wrow0+j, lanes16-31 handle wrow0+8+j
    // (matches the C-matrix row mapping, so embsum[j] is already on the right lanes).
    // lane%16 = embedding dim -> each field read is a contiguous 64B request.
    const uint32_t d = (uint32_t)l16;
#pragma unroll
    for (int j = 0; j < 8; ++j) {
        const int lrow = wrow0 + half * 8 + j;
        const int row  = block_row0 + lrow;
        float s = 0.0f, q = 0.0f;
#pragma unroll
        for (int i = 0; i < NFIELD; ++i) {
            // 32-bit element offset (table is 16M elements) -> saddr+voffset form
            const uint32_t idx = (uint32_t)(s_xcat[lrow * NFIELD + i] + OFF[i]);
            const float e = table[idx * (uint32_t)EDIM + d];
            s += e;
            q += e * e;
        }
        const float lin  = red16(s);          // sum_d s_d
        const float itr2 = red16(s * s - q);  // sum_d s_d^2 - sum_{i,d} E^2
        if (l16 == 0 && row < n_rows)
            out[row] = b0 + lin + embsum[j] + 0.5f * itr2;
    }
}

extern "C" void kernel_launch(void* const* d_in, const int* in_sizes, int n_in,
                              void* d_out, int out_size, void* d_ws, size_t ws_size,
                              hipStream_t stream) {
    const int*   x_cat = (const int*)  d_in[0];
    const float* x_num = (const float*)d_in[1];
    const float* table = (const float*)d_in[2];
    const float* W_num = (const float*)d_in[3];
    const float* bias  = (const float*)d_in[4];
    float* out = (float*)d_out;

    const int n_rows = in_sizes[0] / NFIELD;           // 262144
    const int grid   = (n_rows + ROWS_PER_BLOCK - 1) / ROWS_PER_BLOCK;

    hipLaunchKernelGGL(FieldAwareFM_kernel, dim3(grid), dim3(THREADS), 0, stream,
                       x_cat, x_num, table, W_num, bias, out, n_rows);
}